// EfficientMemorySiLU_37580963840694
// MI455X (gfx1250) — compile-verified
//
#include <hip/hip_runtime.h>
#include <hip/hip_bf16.h>

// SiLU forward: out = x * sigmoid(x), fp32 in / fp32 out. (4,4096,4096) = 67.1M elems.
// HBM-bound: 537 MB @ 23.3 TB/s ~= 23 us floor. Strategy:
//   - one contiguous 16KB tile per block (1024 float4), no loop in main kernel
//   - uniform block base -> SGPR pair; per-thread offset = tid*16 (one v_lshlrev_b32)
//   - 4 loads/stores per thread differ only in 24-bit IOFFSET -> pure
//     global_load_b128/global_store_b128 saddr+voffset+imm, zero per-load VALU
//   - non-temporal hints both ways (stream >> 192MB L2), global_prefetch_b8 ahead
//   - v_exp_f32 / v_rcp_f32 fast path: silu(x) = x * rcp(1 + exp2(-x*log2e))

typedef float v4f __attribute__((ext_vector_type(4)));

#define THREADS 256
#define VEC_PER_THREAD 4
#define VEC_PER_BLOCK (THREADS * VEC_PER_THREAD)   // 1024 float4 = 16KB per block

__device__ __forceinline__ float silu_fast(float t) {
    // exp(-t) = exp2(-t * log2(e)); v_exp_f32 computes 2^x natively.
    float e = __builtin_amdgcn_exp2f(t * -1.44269504088896340736f);
    // x / (1 + e^{-x}) via v_rcp_f32 (~1 ulp, fine for SiLU).
    return t * __builtin_amdgcn_rcpf(1.0f + e);
}

__device__ __forceinline__ v4f silu4(v4f v) {
    v4f r;
    r.x = silu_fast(v.x);
    r.y = silu_fast(v.y);
    r.z = silu_fast(v.z);
    r.w = silu_fast(v.w);
    return r;
}

// Full tiles only: every block processes exactly VEC_PER_BLOCK float4s.
__global__ void __launch_bounds__(THREADS)
silu_fwd_main(const float* __restrict__ x, float* __restrict__ out) {
    // Uniform 64-bit block base -> scalar regs; thread offset stays 32-bit.
    const v4f* __restrict__ bx =
        (const v4f*)x + (unsigned long long)blockIdx.x * VEC_PER_BLOCK;
    v4f* __restrict__ bo =
        (v4f*)out + (unsigned long long)blockIdx.x * VEC_PER_BLOCK;

    unsigned int t = threadIdx.x;

    // Speculative prefetch 8 tiles (128KB) ahead; dropped silently if OOB.
    __builtin_prefetch(&bx[t + 8 * VEC_PER_BLOCK], 0, 0);

    // 4 independent B128 NT loads, immediate offsets 0/4096/8192/12288 bytes.
    v4f a = __builtin_nontemporal_load(&bx[t]);
    v4f b = __builtin_nontemporal_load(&bx[t + 1 * THREADS]);
    v4f c = __builtin_nontemporal_load(&bx[t + 2 * THREADS]);
    v4f d = __builtin_nontemporal_load(&bx[t + 3 * THREADS]);

    v4f ra = silu4(a);
    v4f rb = silu4(b);
    v4f rc = silu4(c);
    v4f rd = silu4(d);

    __builtin_nontemporal_store(ra, &bo[t]);
    __builtin_nontemporal_store(rb, &bo[t + 1 * THREADS]);
    __builtin_nontemporal_store(rc, &bo[t + 2 * THREADS]);
    __builtin_nontemporal_store(rd, &bo[t + 3 * THREADS]);
}

// Scalar grid-stride tail for any remainder (not hit for 4*4096*4096).
__global__ void __launch_bounds__(64)
silu_fwd_tail(const float* __restrict__ x, float* __restrict__ out,
              long long start, long long n) {
    long long i = start + (long long)blockIdx.x * blockDim.x + threadIdx.x;
    long long stride = (long long)gridDim.x * blockDim.x;
    for (; i < n; i += stride) {
        out[i] = silu_fast(x[i]);
    }
}

extern "C" void kernel_launch(void* const* d_in, const int* in_sizes, int n_in,
                              void* d_out, int out_size, void* d_ws, size_t ws_size,
                              hipStream_t stream) {
    (void)n_in; (void)d_ws; (void)ws_size; (void)out_size;

    const float* x = (const float*)d_in[0];
    float* out = (float*)d_out;

    long long n = (long long)in_sizes[0];                       // 67,108,864
    long long elems_per_block = (long long)VEC_PER_BLOCK * 4;   // 4096 floats
    long long full_blocks = n / elems_per_block;                // 16384 for ref size

    if (full_blocks > 0) {
        silu_fwd_main<<<(int)full_blocks, THREADS, 0, stream>>>(x, out);
    }

    long long done = full_blocks * elems_per_block;
    long long rem = n - done;
    if (rem > 0) {
        int tblocks = (int)((rem + 63) / 64);
        if (tblocks > 1024) tblocks = 1024;
        silu_fwd_tail<<<tblocks, 64, 0, stream>>>(x, out, done, n);
    }
}